// GlobalGraph_20899310862590
// MI455X (gfx1250) — compile-verified
//
#include <hip/hip_runtime.h>
#include <math.h>
#include <stdint.h>

// ---------------------------------------------------------------------------
// Masked self-attention with QKV projection, fp32, flash-style fusion.
// - All GEMMs on V_WMMA_F32_16X16X4_F32 (CDNA5 wave32 matrix pipe).
// - K/V tiles staged into LDS via GLOBAL_LOAD_ASYNC_TO_LDS_B128 (ASYNCcnt),
//   shared by all 4 waves of the block (4x global-traffic reduction).
// - W (projection) transposed-staged in LDS: conflict-free ds_load_b64 frags.
// - S accumulation uses 4 independent XDL chains to hide WMMA latency.
// ---------------------------------------------------------------------------

typedef __attribute__((ext_vector_type(2))) float v2f;
typedef __attribute__((ext_vector_type(4))) float v4f;
typedef __attribute__((ext_vector_type(8))) float v8f;

#define B_ 32
#define N_ 1024
#define C_ 256
#define W_ 256
#define NEG_ (-1.0e12f)

#define PAD_KV 260   // row stride for K/V tiles in LDS (mult of 4, conflict-light)
#define PAD_P  18    // row stride for P tile (even -> b64-aligned)
#define PAD_W  34    // col stride for transposed W chunk (conflict-free b64 reads)

static __device__ __forceinline__ v8f wmma_f32(v2f a, v2f b, v8f c) {
  // 8 args: (neg_a, A, neg_b, B, c_mod, C, reuse_a, reuse_b)
  return __builtin_amdgcn_wmma_f32_16x16x4_f32(false, a, false, b,
                                               (short)0, c, false, false);
}

// One-instruction async DMA of 16 bytes/lane from global to LDS (ASYNCcnt).
// LDS destination address = low 32 bits of the generic pointer (flat-aperture
// rule: workgroup-relative — same value DS ops use).
static __device__ __forceinline__ void async_g2l_b128(void* lds_dst,
                                                      const void* gsrc) {
  const unsigned ldsaddr = (unsigned)(uintptr_t)lds_dst;
  asm volatile("global_load_async_to_lds_b128 %0, %1, off"
               :: "v"(ldsaddr), "v"(gsrc) : "memory");
}
static __device__ __forceinline__ void wait_asynccnt0() {
  asm volatile("s_wait_asynccnt 0x0" ::: "memory");
}

// ---------------------------------------------------------------------------
// Kernel 1: Q/K/V = x @ W + b   (x: 32768x256, W: 256x256)
// grid = (512 row-blocks, 3 projections), block = 128 (4 waves).
// Block covers 64 rows x 256 cols; W is staged in 32-row chunks, TRANSPOSED
// into LDS so B-fragments are single conflict-free ds_load_b64s.
// ---------------------------------------------------------------------------
__global__ void __launch_bounds__(128) qkv_proj_kernel(
    const float* __restrict__ x,
    const float* __restrict__ Wq, const float* __restrict__ bq,
    const float* __restrict__ Wk, const float* __restrict__ bk,
    const float* __restrict__ Wv, const float* __restrict__ bv,
    float* __restrict__ q, float* __restrict__ k, float* __restrict__ v)
{
  __shared__ float ldsW[W_ * PAD_W];   // ldsW[col*34 + krel] = W[kc+krel][col]

  const int tid  = threadIdx.x;
  const int lane = tid & 31;
  const int wave = tid >> 5;
  const int half = lane >> 4;   // which 16-lane half
  const int nc   = lane & 15;   // row (A) / col (B,C) within tile

  const int r0 = blockIdx.x * 64 + wave * 16;
  const int p  = blockIdx.y;

  const float* Wm = (p == 0) ? Wq : (p == 1) ? Wk : Wv;
  const float* bm = (p == 0) ? bq : (p == 1) ? bk : bv;
  float*       om = (p == 0) ? q  : (p == 1) ? k  : v;

  v8f acc[16];
#pragma unroll
  for (int t = 0; t < 16; ++t) acc[t] = (v8f){};

  // A fragment (16x4 f32): lane holds x[r0+nc][k + half*2 + {0,1}]
  const float* arow = x + (size_t)(r0 + nc) * C_ + half * 2;

  for (int kc = 0; kc < C_; kc += 32) {
    __syncthreads();   // previous chunk's LDS reads complete

    // cooperative transpose-stage: W[kc..kc+31][0..255] -> ldsW[col][krel]
    // (element-wise scatter -> must go through VGPRs, not async DMA)
#pragma unroll
    for (int rep = 0; rep < 16; ++rep) {
      const int idx = rep * 128 + tid;   // 0..2047 float4 groups
      const int kr  = idx >> 6;          // 0..31
      const int c4  = (idx & 63) << 2;   // 0..252
      v4f w = *(const v4f*)(Wm + (size_t)(kc + kr) * W_ + c4);
      ldsW[(c4 + 0) * PAD_W + kr] = w.x;
      ldsW[(c4 + 1) * PAD_W + kr] = w.y;
      ldsW[(c4 + 2) * PAD_W + kr] = w.z;
      ldsW[(c4 + 3) * PAD_W + kr] = w.w;
    }
    __syncthreads();

#pragma unroll
    for (int kr = 0; kr < 32; kr += 4) {
      v2f a = *(const v2f*)(arow + kc + kr);
#pragma unroll
      for (int t = 0; t < 16; ++t) {
        // B[k][n] = W[kc+kr+half*2 +{0,1}][t*16+nc], one b64 LDS read
        v2f bf = *(const v2f*)(&ldsW[(t * 16 + nc) * PAD_W + kr + half * 2]);
        acc[t] = wmma_f32(a, bf, acc[t]);
      }
    }
  }

  // C/D layout: VGPR i -> row half*8+i, col nc
#pragma unroll
  for (int t = 0; t < 16; ++t) {
    const int col  = t * 16 + nc;
    const float bb = bm[col];
#pragma unroll
    for (int i = 0; i < 8; ++i) {
      const int row = r0 + half * 8 + i;
      om[(size_t)row * W_ + col] = acc[t][i] + bb;
    }
  }
}

// ---------------------------------------------------------------------------
// Kernel 2: flash attention with length masking.
// grid = (16 rowgroups, 32 batches), block = 128 (4 waves, same batch).
// K/V blocks staged per m-iteration via async global->LDS DMA and shared by
// all 4 waves. S uses 4 independent WMMA chains.
// ---------------------------------------------------------------------------
__global__ void __launch_bounds__(128) attention_kernel(
    const float* __restrict__ q, const float* __restrict__ k,
    const float* __restrict__ v, const int* __restrict__ valid_lens,
    float* __restrict__ out)
{
  __shared__ float ldsK[16 * PAD_KV];
  __shared__ float ldsV[16 * PAD_KV];
  __shared__ float ldsP[4][16 * PAD_P];   // per-wave 16x16 P tile

  const int tid  = threadIdx.x;
  const int lane = tid & 31;
  const int wave = tid >> 5;
  const int half = lane >> 4;
  const int nc   = lane & 15;

  const int b  = blockIdx.y;
  const int rb = blockIdx.x * 4 + wave;   // 64 row-tiles per batch
  const int r0 = rb * 16;
  const int vl = valid_lens[b];

  const float* qb = q + (size_t)b * N_ * W_;
  const float* kb = k + (size_t)b * N_ * W_;
  const float* vb = v + (size_t)b * N_ * W_;

  v8f o[16];
#pragma unroll
  for (int t = 0; t < 16; ++t) o[t] = (v8f){};

  float rmax[8], rsum[8];
#pragma unroll
  for (int i = 0; i < 8; ++i) { rmax[i] = -3.0e38f; rsum[i] = 0.0f; }

  const float* qrow = qb + (size_t)(r0 + nc) * W_ + half * 2;
  float* Pw = &ldsP[wave][0];

  // this thread's slice of the cooperative K/V stage (8 float4s per tile)
  const int srow = tid >> 3;          // 0..15 (row handled by this thread)
  const int sc4  = (tid & 7) << 5;    // 0,32,..,224 (four float4s per rep)

  for (int m0 = 0; m0 < N_; m0 += 16) {
    if (m0 >= vl) break;  // uniform across block (all waves share batch b)

    __syncthreads();   // previous iteration's ldsK/ldsV reads complete

    // ---- async DMA stage of K/V blocks (16 x 256 floats each) ----
    // 128 threads x (4+4) b128 transfers x 2 reps = 2 x 16KB
#pragma unroll
    for (int rep = 0; rep < 2; ++rep) {
      const int c4 = sc4 + rep * 16;   // float index, 16-float steps
      async_g2l_b128(&ldsK[srow * PAD_KV + c4 + 0],
                     kb + (size_t)(m0 + srow) * W_ + c4 + 0);
      async_g2l_b128(&ldsK[srow * PAD_KV + c4 + 4],
                     kb + (size_t)(m0 + srow) * W_ + c4 + 4);
      async_g2l_b128(&ldsV[srow * PAD_KV + c4 + 0],
                     vb + (size_t)(m0 + srow) * W_ + c4 + 0);
      async_g2l_b128(&ldsV[srow * PAD_KV + c4 + 4],
                     vb + (size_t)(m0 + srow) * W_ + c4 + 4);
      async_g2l_b128(&ldsK[srow * PAD_KV + c4 + 8],
                     kb + (size_t)(m0 + srow) * W_ + c4 + 8);
      async_g2l_b128(&ldsK[srow * PAD_KV + c4 + 12],
                     kb + (size_t)(m0 + srow) * W_ + c4 + 12);
      async_g2l_b128(&ldsV[srow * PAD_KV + c4 + 8],
                     vb + (size_t)(m0 + srow) * W_ + c4 + 8);
      async_g2l_b128(&ldsV[srow * PAD_KV + c4 + 12],
                     vb + (size_t)(m0 + srow) * W_ + c4 + 12);
    }
    wait_asynccnt0();
    __syncthreads();

    // ---- S = Q @ K^T : 4 independent accumulator chains ----
    v8f s0 = {}, s1 = {}, s2 = {}, s3 = {};
    const float* kld = &ldsK[nc * PAD_KV + half * 2];
#pragma unroll 4
    for (int k0 = 0; k0 < C_; k0 += 16) {
      v2f a0 = *(const v2f*)(qrow + k0);
      v2f a1 = *(const v2f*)(qrow + k0 + 4);
      v2f a2 = *(const v2f*)(qrow + k0 + 8);
      v2f a3 = *(const v2f*)(qrow + k0 + 12);
      v2f b0 = *(const v2f*)(kld + k0);
      v2f b1 = *(const v2f*)(kld + k0 + 4);
      v2f b2 = *(const v2f*)(kld + k0 + 8);
      v2f b3 = *(const v2f*)(kld + k0 + 12);
      s0 = wmma_f32(a0, b0, s0);
      s1 = wmma_f32(a1, b1, s1);
      s2 = wmma_f32(a2, b2, s2);
      s3 = wmma_f32(a3, b3, s3);
    }
    v8f s = (s0 + s1) + (s2 + s3);

    const bool colmask = ((m0 + nc) >= vl);

    float pv[8], corr[8];
#pragma unroll
    for (int i = 0; i < 8; ++i) {
      // scale by 1/sqrt(W)=1/16, then mask (matches reference order)
      float sv = colmask ? NEG_ : s[i] * 0.0625f;

      // row max over the 16 columns held by this half (xor<16 stays in half)
      float tm = sv;
      tm = fmaxf(tm, __shfl_xor(tm, 1, 32));
      tm = fmaxf(tm, __shfl_xor(tm, 2, 32));
      tm = fmaxf(tm, __shfl_xor(tm, 4, 32));
      tm = fmaxf(tm, __shfl_xor(tm, 8, 32));

      const float newmax = fmaxf(rmax[i], tm);
      corr[i] = __expf(rmax[i] - newmax);
      rmax[i] = newmax;

      const float pe = __expf(sv - newmax);  // masked cols underflow to 0
      pv[i] = pe;

      float ss = pe;
      ss += __shfl_xor(ss, 1, 32);
      ss += __shfl_xor(ss, 2, 32);
      ss += __shfl_xor(ss, 4, 32);
      ss += __shfl_xor(ss, 8, 32);
      rsum[i] = rsum[i] * corr[i] + ss;
    }

    // rescale running output (row i of each d-tile)
#pragma unroll
    for (int t = 0; t < 16; ++t) {
#pragma unroll
      for (int i = 0; i < 8; ++i) o[t][i] *= corr[i];
    }

    // C-layout -> per-wave LDS tile (row-major 16x16, stride 18);
    // same-wave DS ops are in-order, so no extra barrier needed for P
#pragma unroll
    for (int i = 0; i < 8; ++i)
      Pw[(half * 8 + i) * PAD_P + nc] = pv[i];

    // ---- O += P @ V : A from LDS P (transposed view), B from LDS V ----
#pragma unroll
    for (int kk = 0; kk < 4; ++kk) {
      v2f aP = *(const v2f*)(Pw + nc * PAD_P + kk * 4 + half * 2);
      const int rbase = kk * 4 + half * 2;
#pragma unroll
      for (int t = 0; t < 16; ++t) {
        const int d = t * 16 + nc;
        v2f bf;
        bf.x = ldsV[(rbase + 0) * PAD_KV + d];
        bf.y = ldsV[(rbase + 1) * PAD_KV + d];
        o[t] = wmma_f32(aP, bf, o[t]);
      }
    }
  }

  // ---- normalize + store; fully-masked rows (row >= vl) output 0 ----
  float* ob = out + (size_t)b * N_ * W_;
#pragma unroll
  for (int i = 0; i < 8; ++i) {
    const int row = r0 + half * 8 + i;
    const float inv = (row < vl) ? (1.0f / rsum[i]) : 0.0f;
#pragma unroll
    for (int t = 0; t < 16; ++t)
      ob[(size_t)row * W_ + t * 16 + nc] = o[t][i] * inv;
  }
}

// ---------------------------------------------------------------------------
extern "C" void kernel_launch(void* const* d_in, const int* in_sizes, int n_in,
                              void* d_out, int out_size, void* d_ws, size_t ws_size,
                              hipStream_t stream) {
  const float* x  = (const float*)d_in[0];
  const int*   vl = (const int*)d_in[1];
  const float* Wq = (const float*)d_in[2];
  const float* bq = (const float*)d_in[3];
  const float* Wk = (const float*)d_in[4];
  const float* bk = (const float*)d_in[5];
  const float* Wv = (const float*)d_in[6];
  const float* bv = (const float*)d_in[7];
  float* out = (float*)d_out;

  const size_t qkv_elems = (size_t)B_ * N_ * W_;   // 8388608
  float* q = (float*)d_ws;
  float* k = q + qkv_elems;
  float* v = k + qkv_elems;

  dim3 g1((B_ * N_) / 64, 3);   // (512, 3)
  qkv_proj_kernel<<<g1, 128, 0, stream>>>(x, Wq, bq, Wk, bk, Wv, bv, q, k, v);

  dim3 g2((N_ / 16) / 4, B_);   // (16, 32)
  attention_kernel<<<g2, 128, 0, stream>>>(q, k, v, vl, out);
}